// HomoHingeLoss_3393024164480
// MI455X (gfx1250) — compile-verified
//
#include <hip/hip_runtime.h>

// ---- problem constants (from reference) ----
#define GS   8
#define HRR  60
#define WRR  60
#define PP   (HRR * WRR)   // 3600 spatial positions per batch
#define CC   256           // channels (GEMM K)
#define NB   4             // batch
#define HH   (HRR * GS)    // 480
#define WW   (WRR * GS)    // 480

// ---- tiling ----
#define BM   128
#define BN   128
#define KB   32
#define LDP  136           // LDS row stride (136 % 64 == 8 -> conflict-free frag loads)

#define HAS_ASYNC_LDS (__has_builtin(__builtin_amdgcn_global_load_async_to_lds_b128))

typedef __attribute__((ext_vector_type(2))) float v2f;
typedef __attribute__((ext_vector_type(8))) float v8f;

// GCC-style vector types matching the async-LDS builtin prototype
typedef int v4i_b __attribute__((vector_size(16)));
typedef __attribute__((address_space(1))) v4i_b gas_v4i;   // global (AS1)
typedef __attribute__((address_space(3))) v4i_b las_v4i;   // LDS (AS3)

// ---------------------------------------------------------------------------
// Prep: per (n,q) warped grid point, 8x8 visibility product, vm_sum reduction
// ---------------------------------------------------------------------------
__global__ __launch_bounds__(256) void hh_prep_kernel(
    const float* __restrict__ homo, const float* __restrict__ vis,
    float* __restrict__ vm, float* __restrict__ wg, float* __restrict__ vm_sum) {
  const int n = blockIdx.y;
  const int q = blockIdx.x * 256 + threadIdx.x;
  float prod = 0.0f;
  if (q < PP) {
    const int h = q / WRR;
    const int w = q - h * WRR;
    const float gx = (float)(w * GS + GS / 2);
    const float gy = (float)(h * GS + GS / 2);
    const float* Hm = homo + n * 9;
    const float X = Hm[0] * gx + Hm[1] * gy + Hm[2];
    const float Y = Hm[3] * gx + Hm[4] * gy + Hm[5];
    const float Z = Hm[6] * gx + Hm[7] * gy + Hm[8];
    const float iz = 1.0f / Z;
    wg[(n * PP + q) * 2 + 0] = X * iz;
    wg[(n * PP + q) * 2 + 1] = Y * iz;
    prod = 1.0f;
    const float* vp = vis + (size_t)n * HH * WW + (size_t)(h * GS) * WW + w * GS;
    #pragma unroll
    for (int yy = 0; yy < GS; ++yy)
      #pragma unroll
      for (int xx = 0; xx < GS; ++xx)
        prod *= vp[yy * WW + xx];
    vm[n * PP + q] = prod;
  }
  __shared__ float red[256];
  red[threadIdx.x] = prod;
  __syncthreads();
  for (int s = 128; s > 0; s >>= 1) {
    if (threadIdx.x < s) red[threadIdx.x] += red[threadIdx.x + s];
    __syncthreads();
  }
  if (threadIdx.x == 0) atomicAdd(vm_sum, red[0]);
}

// ---------------------------------------------------------------------------
// Stage one 32(K) x 128(P) f32 block into LDS.
// Async path: GLOBAL_LOAD_ASYNC_TO_LDS_B128 (ASYNCcnt tracked), addresses
// clamped into the valid row for partial tiles (masked in epilogue).
// ---------------------------------------------------------------------------
__device__ __forceinline__ void hh_stage_block(const float* __restrict__ g0,
                                               float* __restrict__ lbuf,
                                               int kb, int p0, int tid) {
  #pragma unroll
  for (int i = 0; i < 4; ++i) {
    const int f   = tid + i * 256;
    const int row = f >> 5;          // 0..31 (K within block)
    const int col = (f & 31) << 2;   // 0..124
    int p = p0 + col;
    if (p > PP - 4) p = PP - 4;      // clamp (dup data only feeds masked outputs)
    const float* src = g0 + (size_t)(kb + row) * PP + p;
    float* dst = lbuf + row * LDP + col;
#if HAS_ASYNC_LDS
    __builtin_amdgcn_global_load_async_to_lds_b128(
        (gas_v4i*)src, (las_v4i*)dst, 0, 0);
#else
    *(float4*)dst = *(const float4*)src;
#endif
  }
}

__device__ __forceinline__ void hh_wait_stage() {
#if HAS_ASYNC_LDS
#if __has_builtin(__builtin_amdgcn_s_wait_asynccnt)
  __builtin_amdgcn_s_wait_asynccnt(0);
#else
  asm volatile("s_wait_asynccnt 0" ::: "memory");
#endif
#endif
  __syncthreads();
}

// ---------------------------------------------------------------------------
// Fused GEMM (f32 WMMA 16x16x4, double-buffered async LDS staging)
// + hinge-loss epilogue + global reduction.
// D[p,q] = sum_c desc1[n,c,p]*desc2[n,c,q]; loss folded per element.
// ---------------------------------------------------------------------------
__global__ __launch_bounds__(256) void hh_gemm_loss_kernel(
    const float* __restrict__ d1, const float* __restrict__ d2,
    const float* __restrict__ vm, const float* __restrict__ wg,
    float* __restrict__ loss_sum) {
  __shared__ float bufA[2][KB * LDP];
  __shared__ float bufB[2][KB * LDP];
  __shared__ float red[256];

  const int tid = threadIdx.x;
  const int n   = blockIdx.z;
  const int pm0 = blockIdx.x * BM;   // desc1 (M) tile origin
  const int pn0 = blockIdx.y * BN;   // desc2 (N) tile origin

  const int wave = tid >> 5;
  const int lane = tid & 31;
  const int wm = wave >> 2;          // 0..1 : 64-row half of macro tile
  const int wn = wave & 3;           // 0..3 : 32-col quarter
  const int ml = lane & 15;
  const int hi = lane >> 4;          // half-wave select
  const int ko = hi << 1;            // K sub-offset 0 or 2 (ISA f32 A/B layout)

  v8f acc[4][2];
  #pragma unroll
  for (int i = 0; i < 4; ++i)
    #pragma unroll
    for (int j = 0; j < 2; ++j) acc[i][j] = (v8f)0.0f;

  const float* A0 = d1 + (size_t)n * CC * PP;
  const float* B0 = d2 + (size_t)n * CC * PP;

  // pre-issue first K-block into buffer 0
  hh_stage_block(A0, &bufA[0][0], 0, pm0, tid);
  hh_stage_block(B0, &bufB[0][0], 0, pn0, tid);

  for (int kb = 0; kb < CC; kb += KB) {
    const int cur = (kb >> 5) & 1;
    hh_wait_stage();                         // own asyncs done + all waves synced
    if (kb + KB < CC) {                      // prefetch next block into other buf
      hh_stage_block(A0, &bufA[cur ^ 1][0], kb + KB, pm0, tid);
      hh_stage_block(B0, &bufB[cur ^ 1][0], kb + KB, pn0, tid);
    }
    const float* lA = &bufA[cur][0];
    const float* lB = &bufB[cur][0];

    #pragma unroll
    for (int kk = 0; kk < KB; kk += 4) {
      const int r0 = (kk + ko) * LDP;
      const int r1 = r0 + LDP;
      v2f afr[4], bfr[2];
      #pragma unroll
      for (int i = 0; i < 4; ++i) {
        const int cA = wm * 64 + i * 16 + ml;
        afr[i].x = lA[r0 + cA];
        afr[i].y = lA[r1 + cA];
      }
      #pragma unroll
      for (int j = 0; j < 2; ++j) {
        const int cB = wn * 32 + j * 16 + ml;
        bfr[j].x = lB[r0 + cB];
        bfr[j].y = lB[r1 + cB];
      }
      #pragma unroll
      for (int i = 0; i < 4; ++i)
        #pragma unroll
        for (int j = 0; j < 2; ++j)
          acc[i][j] = __builtin_amdgcn_wmma_f32_16x16x4_f32(
              false, afr[i], false, bfr[j], (short)0, acc[i][j], false, false);
    }
  }

  // ---- fused hinge-loss epilogue ----
  float lsum = 0.0f;
  #pragma unroll
  for (int j = 0; j < 2; ++j) {
    const int q = pn0 + wn * 32 + j * 16 + ml;  // desc2 position (fixed per lane)
    if (q < PP) {
      const float wx  = wg[(n * PP + q) * 2 + 0];
      const float wy  = wg[(n * PP + q) * 2 + 1];
      const float vmq = vm[n * PP + q];
      #pragma unroll
      for (int i = 0; i < 4; ++i) {
        const int mb = pm0 + wm * 64 + i * 16 + hi * 8;
        #pragma unroll
        for (int r = 0; r < 8; ++r) {
          const int m = mb + r;
          if (m < PP) {
            const int h = m / WRR;
            const int w = m - h * WRR;
            const float gx = (float)(w * GS + GS / 2);
            const float gy = (float)(h * GS + GS / 2);
            const float dx = gx - wx, dy = gy - wy;
            const float dot = acc[i][j][r];
            const float pos = fmaxf(1.0f - dot, 0.0f);   // POS_MARGIN hinge
            const float neg = fmaxf(dot - 0.2f, 0.0f);   // NEG_MARGIN hinge
            // s = (dist <= 7.5) <=> dist^2 <= 56.25 ; both branches scaled by vm
            lsum += vmq * ((dx * dx + dy * dy <= 56.25f) ? pos : neg);
          }
        }
      }
    }
  }
  red[tid] = lsum;
  __syncthreads();
  for (int s = 128; s > 0; s >>= 1) {
    if (tid < s) red[tid] += red[tid + s];
    __syncthreads();
  }
  if (tid == 0) atomicAdd(loss_sum, red[0]);
}

// ---------------------------------------------------------------------------
__global__ void hh_finalize_kernel(const float* __restrict__ sums,
                                   float* __restrict__ out) {
  // norm = hr*wr * vm_sum ; LOSS_LAMBDA = 1
  out[0] = sums[0] / ((float)PP * sums[1]);
}

// ---------------------------------------------------------------------------
extern "C" void kernel_launch(void* const* d_in, const int* in_sizes, int n_in,
                              void* d_out, int out_size, void* d_ws, size_t ws_size,
                              hipStream_t stream) {
  const float* desc1 = (const float*)d_in[0];
  const float* desc2 = (const float*)d_in[1];
  const float* homo  = (const float*)d_in[2];
  const float* vis   = (const float*)d_in[3];
  float* out = (float*)d_out;

  float* ws   = (float*)d_ws;
  float* sums = ws;                    // [0]=loss_sum, [1]=vm_sum
  float* vm   = ws + 16;               // NB*PP
  float* wg   = ws + 16 + NB * PP;     // NB*PP*2

  (void)hipMemsetAsync(sums, 0, 2 * sizeof(float), stream);

  dim3 pgrid((PP + 255) / 256, NB);
  hh_prep_kernel<<<pgrid, dim3(256), 0, stream>>>(homo, vis, vm, wg, sums + 1);

  dim3 ggrid((PP + BM - 1) / BM, (PP + BN - 1) / BN, NB);
  hh_gemm_loss_kernel<<<ggrid, dim3(256), 0, stream>>>(desc1, desc2, vm, wg, sums);

  hh_finalize_kernel<<<1, 1, 0, stream>>>(sums, out);
}